// SelfAttention_44444321579182
// MI455X (gfx1250) — compile-verified
//
#include <hip/hip_runtime.h>

#define BB  8
#define LL  64
#define HH  8
#define DD  32
#define HID 256

typedef float v2f __attribute__((ext_vector_type(2)));
typedef float v8f __attribute__((ext_vector_type(8)));
typedef unsigned int u32x4 __attribute__((ext_vector_type(4)));
typedef int i32x4 __attribute__((ext_vector_type(4)));
typedef int i32x8 __attribute__((ext_vector_type(8)));

__device__ __forceinline__ v8f wmma4(v2f a, v2f b, v8f c) {
  // D = A(16x4,f32) * B(4x16,f32) + C(16x16,f32)
  return __builtin_amdgcn_wmma_f32_16x16x4_f32(false, a, false, b, (short)0, c, false, false);
}

// ---------------------------------------------------------------------------
// TDM: async-DMA one contiguous 2048-float (8 KB) row from global into LDS.
// 1-D tensor descriptor per CDNA5 ISA ch.8: group0 {count=1, lds_addr,
// global_addr[56:0], type=2}; group1 {data_size=4B, tensor_dim0=2048,
// tile_dim0=2048, tile_dim1=1, dim0_stride=2048}; groups 2/3 unused (zero).
// Tracked with TENSORcnt; issued by one wave per workgroup (EXEC ignored).
// This toolchain exposes the 6-arg builtin (extra int32x8 group, then cpol).
// ---------------------------------------------------------------------------
__device__ __forceinline__ void tdm_load_row(const float* gsrc, unsigned lds_byte) {
  unsigned long long ga = (unsigned long long)(uintptr_t)gsrc;
  u32x4 g0;
  g0.x = 1u;                                        // count=1 (valid), user mode
  g0.y = lds_byte;                                  // LDS dest byte address
  g0.z = (unsigned)(ga & 0xffffffffu);              // global addr [31:0]
  g0.w = (unsigned)((ga >> 32) & 0x01ffffffu) | 0x80000000u; // [56:32] | type=2
  i32x8 g1;
  g1[0] = 0x00020000;                               // data_size = 2 (4 bytes)
  g1[1] = (int)(2048u << 16);                       // tensor_dim0 = 2048 (lo16)
  g1[2] = (int)(1u << 16);                          // tensor_dim0 hi=0, tensor_dim1 = 1
  g1[3] = (int)(2048u << 16);                       // tile_dim0 = 2048
  g1[4] = 1;                                        // tile_dim1 = 1, tile_dim2 = 0
  g1[5] = 2048;                                     // tensor_dim0_stride = 2048
  g1[6] = 0;
  g1[7] = 0;
  i32x4 g2 = {0, 0, 0, 0};
  i32x4 g3 = {0, 0, 0, 0};
  i32x8 g4 = {0, 0, 0, 0, 0, 0, 0, 0};
  __builtin_amdgcn_tensor_load_to_lds(g0, g1, g2, g3, g4, 0);
}

// ---------------------------------------------------------------------------
// Kernel 1: q/k/v projections + per-head precompute G = Wtk^T k, c2 = btk.k
// ---------------------------------------------------------------------------
__global__ void k_proj_qkv(const float* __restrict__ hid,
                           const float* __restrict__ Wq, const float* __restrict__ bq,
                           const float* __restrict__ Wk, const float* __restrict__ bk,
                           const float* __restrict__ Wv, const float* __restrict__ bv,
                           const float* __restrict__ Wtk, const float* __restrict__ btk,
                           float* __restrict__ qo, float* __restrict__ ko,
                           float* __restrict__ vo, float* __restrict__ gko,
                           float* __restrict__ c2o) {
  __shared__ __align__(16) float hrow[HID];
  __shared__ __align__(16) float klds[HID];
  const int tok = blockIdx.x;            // b*L + i
  const int b = tok / LL, i = tok % LL;
  const int o = threadIdx.x;             // 0..255
  hrow[o] = hid[tok * HID + o];
  __syncthreads();
  float aq = bq[o], ak = bk[o], av = bv[o];
  for (int c = 0; c < HID; ++c) {
    float x = hrow[c];
    aq = fmaf(Wq[o * HID + c], x, aq);
    ak = fmaf(Wk[o * HID + c], x, ak);
    av = fmaf(Wv[o * HID + c], x, av);
  }
  const int h = o >> 5, d = o & 31;
  const int idx = ((b * HH + h) * LL + i) * DD + d;
  qo[idx] = aq; ko[idx] = ak; vo[idx] = av;
  klds[o] = ak;
  __syncthreads();
  // gk[b,h,i,e] = sum_d Wtk[d][e] * k[b,h,i,d]   (e == o&31 here)
  float g = 0.f;
  for (int d2 = 0; d2 < DD; ++d2)
    g = fmaf(Wtk[d2 * DD + d], klds[h * DD + d2], g);
  gko[idx] = g;
  if (d == 0) {
    float c2 = 0.f;
    for (int d2 = 0; d2 < DD; ++d2)
      c2 = fmaf(btk[d2], klds[h * DD + d2], c2);
    c2o[(b * HH + h) * LL + i] = c2;
  }
}

// ---------------------------------------------------------------------------
// Kernel 2: sq / sv projections from structure_matrix
// ---------------------------------------------------------------------------
__global__ void k_proj_sqsv(const float* __restrict__ sm,
                            const float* __restrict__ Wsq, const float* __restrict__ bsq,
                            const float* __restrict__ Wsv, const float* __restrict__ bsv,
                            float* __restrict__ sqo, float* __restrict__ svo) {
  __shared__ __align__(16) float row[HID];
  const int p = blockIdx.x;              // (b*L + i)*L + j
  const int t = threadIdx.x;             // 0..63
  ((float4*)row)[t] = ((const float4*)(sm + (size_t)p * HID))[t];
  __syncthreads();
  const float* W    = (t < DD) ? Wsq : Wsv;
  const float* bias = (t < DD) ? bsq : bsv;
  float*       outp = (t < DD) ? sqo : svo;
  const int oo = t & 31;
  float acc = bias[oo];
  for (int c = 0; c < HID; ++c) acc = fmaf(W[oo * HID + c], row[c], acc);
  outp[p * DD + oo] = acc;
}

// ---------------------------------------------------------------------------
// Shared score-stage: fills Kt, Gkl, Q5, U, c1, c2l, mrow, Wt(=Wtq), btql and
// computes Stile[64x64] = Q5[64x32] x Kt^T via V_WMMA_F32_16X16X4_F32.
// ---------------------------------------------------------------------------
__device__ __forceinline__ void score_stage(
    int b, int h, int i,
    const float* __restrict__ qws, const float* __restrict__ kws,
    const float* __restrict__ gkws, const float* __restrict__ sqws,
    const float* __restrict__ c2ws, const float* __restrict__ mask,
    const float* __restrict__ Wtq, const float* __restrict__ btq,
    float* Kt, float* Gkl, float* Q5, float* U, float* Stile,
    float* Wt, float* c1, float* c2l, float* mrow, float* btql) {
  const int tid = threadIdx.x;
  const int bh  = b * HH + h;

  { // bulk loads: K tile, G tile (2048 floats each), Wtq (1024 floats)
    const float4* ksrc = (const float4*)(kws  + bh * LL * DD);
    const float4* gsrc = (const float4*)(gkws + bh * LL * DD);
    ((float4*)Kt )[tid]       = ksrc[tid];
    ((float4*)Kt )[tid + 256] = ksrc[tid + 256];
    ((float4*)Gkl)[tid]       = gsrc[tid];
    ((float4*)Gkl)[tid + 256] = gsrc[tid + 256];
    ((float4*)Wt )[tid]       = ((const float4*)Wtq)[tid];
  }
  if (tid < DD) btql[tid] = btq[tid];
  if (tid < LL) {
    mrow[tid] = mask[b * LL + tid];
    c2l[tid]  = c2ws[bh * LL + tid];
  }
  // Q5[j][d] = q[i,d] + q[j,d] + sq[i,j,d]
  for (int idx = tid; idx < LL * DD; idx += 256) {
    const int j = idx >> 5, d = idx & 31;
    Q5[idx] = qws[(bh * LL + i) * DD + d] + qws[(bh * LL + j) * DD + d]
            + sqws[((b * LL + i) * LL + j) * DD + d];
  }
  __syncthreads();

  // U[j][e] = sum_d Wtq[d][e] * Q5[j][d];   c1[j] = btq . Q5[j]
#pragma unroll
  for (int r = 0; r < 8; ++r) {
    const int idx = tid * 8 + r;
    const int j = idx >> 5, e = idx & 31;
    float a = 0.f;
    for (int d2 = 0; d2 < DD; ++d2) a = fmaf(Wt[d2 * DD + e], Q5[j * DD + d2], a);
    U[idx] = a;
  }
  if (tid < LL) {
    float a = 0.f;
    for (int d2 = 0; d2 < DD; ++d2) a = fmaf(btql[d2], Q5[tid * DD + d2], a);
    c1[tid] = a;
  }
  __syncthreads();

  // WMMA GEMM: Stile[j][k] = sum_d Q5[j][d] * Kt[k][d]
  const int lane = tid & 31, wv = tid >> 5;
  const int klo  = (lane < 16) ? 0 : 2;
  const int mo   = (lane < 16) ? 0 : 8;
  const int ln   = lane & 15;
  for (int t = wv; t < 16; t += 8) {
    const int j0 = (t >> 2) * 16, k0 = (t & 3) * 16;
    v8f acc = {};
#pragma unroll
    for (int dd2 = 0; dd2 < DD; dd2 += 4) {
      v2f a, bf;
      a.x  = Q5[(j0 + ln) * DD + dd2 + klo];
      a.y  = Q5[(j0 + ln) * DD + dd2 + klo + 1];
      bf.x = Kt[(k0 + ln) * DD + dd2 + klo];
      bf.y = Kt[(k0 + ln) * DD + dd2 + klo + 1];
      acc = wmma4(a, bf, acc);
    }
#pragma unroll
    for (int r = 0; r < 8; ++r)
      Stile[(j0 + r + mo) * LL + (k0 + ln)] = acc[r];
  }
  __syncthreads();
}

#define SCALE 0.17677669529663687f /* 1/sqrt(32) */
#define TRIP_ROW(b, i, j) ((((b) * LL + (i)) * LL + (j)) * LL * DD)

// ---------------------------------------------------------------------------
// Pass 1: S[b,h,k] = sum_{i,j} exp(score). Triple rows streamed by the TDM,
// double-buffered between `tripl` and the dead Q5 region.
// ---------------------------------------------------------------------------
__global__ void k_pass_sum(const float* __restrict__ qws, const float* __restrict__ kws,
                           const float* __restrict__ gkws, const float* __restrict__ sqws,
                           const float* __restrict__ c2ws, const float* __restrict__ mask,
                           const float* __restrict__ triple,
                           const float* __restrict__ Wtq, const float* __restrict__ btq,
                           float* __restrict__ Ssum) {
  __shared__ __align__(16) float Kt[LL * DD];
  __shared__ __align__(16) float Gkl[LL * DD];
  __shared__ __align__(16) float Q5[LL * DD];    // trip buffer 1 after WMMA
  __shared__ __align__(16) float U[LL * DD];
  __shared__ __align__(16) float Stile[LL * LL];
  __shared__ __align__(16) float tripl[LL * DD]; // trip buffer 0
  __shared__ __align__(16) float Wt[DD * DD];
  __shared__ float c1[LL], c2l[LL], mrow[LL], btql[DD];

  const int h = blockIdx.x, i = blockIdx.y, b = blockIdx.z;
  const int tid = threadIdx.x;
  const int bh = b * HH + h;
  const unsigned lds0 = (unsigned)(uintptr_t)tripl;
  const unsigned lds1 = (unsigned)(uintptr_t)Q5;

  if (tid < 32) tdm_load_row(triple + TRIP_ROW(b, i, 0), lds0);  // overlaps stage

  score_stage(b, h, i, qws, kws, gkws, sqws, c2ws, mask, Wtq, btq,
              Kt, Gkl, Q5, U, Stile, Wt, c1, c2l, mrow, btql);

  if (tid < 32) tdm_load_row(triple + TRIP_ROW(b, i, 1), lds1);  // Q5 now dead

  const int kk = tid >> 2, qd = tid & 3;
  const float mi = mrow[i];
  float accsum = 0.f;
  for (int j = 0; j < LL; ++j) {
    if (tid < 32) {
      if (j < LL - 1) __builtin_amdgcn_s_wait_tensorcnt(1);  // row j landed
      else            __builtin_amdgcn_s_wait_tensorcnt(0);
    }
    __syncthreads();
    const float* tb = (j & 1) ? Q5 : tripl;
    float tv = 0.f;
#pragma unroll
    for (int r = 0; r < 8; ++r) {
      const int d2 = qd * 8 + r;
      tv = fmaf(tb[kk * DD + d2], U[j * DD + d2] + Gkl[kk * DD + d2], tv);
    }
    tv += __shfl_xor(tv, 1, 32);
    tv += __shfl_xor(tv, 2, 32);
    if (qd == 0) {
      const float s = (Stile[j * LL + kk] + c1[j] + c2l[kk] + tv) * SCALE
                      + mi * mrow[j] * mrow[kk];
      accsum += expf(s);
    }
    __syncthreads();
    if (tid < 32 && j + 2 < LL)        // refill the buffer just consumed
      tdm_load_row(triple + TRIP_ROW(b, i, j + 2), (j & 1) ? lds1 : lds0);
  }
  if (qd == 0) atomicAdd(&Ssum[bh * LL + kk], accsum);
}

// ---------------------------------------------------------------------------
// Pass 2: recompute scores, normalize, ctx[k,d] += P^T x V5 via WMMA
// ---------------------------------------------------------------------------
__global__ void k_pass_ctx(const float* __restrict__ qws, const float* __restrict__ kws,
                           const float* __restrict__ vws, const float* __restrict__ gkws,
                           const float* __restrict__ sqws, const float* __restrict__ svws,
                           const float* __restrict__ c2ws, const float* __restrict__ mask,
                           const float* __restrict__ triple,
                           const float* __restrict__ Wtq, const float* __restrict__ btq,
                           const float* __restrict__ Ssum, float* __restrict__ out) {
  __shared__ __align__(16) float Kt[LL * DD];
  __shared__ __align__(16) float Gkl[LL * DD];
  __shared__ __align__(16) float Q5[LL * DD];    // trip buffer 1, then V5
  __shared__ __align__(16) float U[LL * DD];
  __shared__ __align__(16) float Stile[LL * LL]; // later reused as P tile
  __shared__ __align__(16) float tripl[LL * DD]; // trip buffer 0
  __shared__ __align__(16) float Wt[DD * DD];
  __shared__ float c1[LL], c2l[LL], mrow[LL], btql[DD], rS[LL];

  const int h = blockIdx.x, i = blockIdx.y, b = blockIdx.z;
  const int tid = threadIdx.x;
  const int bh = b * HH + h;
  const unsigned lds0 = (unsigned)(uintptr_t)tripl;
  const unsigned lds1 = (unsigned)(uintptr_t)Q5;

  if (tid < 32) tdm_load_row(triple + TRIP_ROW(b, i, 0), lds0);
  if (tid < LL) rS[tid] = 1.0f / Ssum[bh * LL + tid];

  score_stage(b, h, i, qws, kws, gkws, sqws, c2ws, mask, Wtq, btq,
              Kt, Gkl, Q5, U, Stile, Wt, c1, c2l, mrow, btql);

  if (tid < 32) tdm_load_row(triple + TRIP_ROW(b, i, 1), lds1);

  const int kk = tid >> 2, qd = tid & 3;
  const float mi = mrow[i];
  for (int j = 0; j < LL; ++j) {
    if (tid < 32) {
      if (j < LL - 1) __builtin_amdgcn_s_wait_tensorcnt(1);
      else            __builtin_amdgcn_s_wait_tensorcnt(0);
    }
    __syncthreads();
    const float* tb = (j & 1) ? Q5 : tripl;
    float tv = 0.f;
#pragma unroll
    for (int r = 0; r < 8; ++r) {
      const int d2 = qd * 8 + r;
      tv = fmaf(tb[kk * DD + d2], U[j * DD + d2] + Gkl[kk * DD + d2], tv);
    }
    tv += __shfl_xor(tv, 1, 32);
    tv += __shfl_xor(tv, 2, 32);
    if (qd == 0) {
      const float s = (Stile[j * LL + kk] + c1[j] + c2l[kk] + tv) * SCALE
                      + mi * mrow[j] * mrow[kk];
      Stile[j * LL + kk] = expf(s) * rS[kk];   // P tile, in place
    }
    __syncthreads();
    if (tid < 32 && j + 2 < LL)
      tdm_load_row(triple + TRIP_ROW(b, i, j + 2), (j & 1) ? lds1 : lds0);
  }

  // V5[j][d] = v[i,d] + v[j,d] + sv[i,j,d]  (reuse Q5 storage; TENSORcnt==0)
  for (int idx = tid; idx < LL * DD; idx += 256) {
    const int j = idx >> 5, d = idx & 31;
    Q5[idx] = vws[(bh * LL + i) * DD + d] + vws[(bh * LL + j) * DD + d]
            + svws[((b * LL + i) * LL + j) * DD + d];
  }
  __syncthreads();

  // ctx[k,d] += sum_j P[j,k] * V5[j,d] : A = P^T (16x4 frags), B = V5
  const int lane = tid & 31, wv = tid >> 5;
  const int klo  = (lane < 16) ? 0 : 2;
  const int mo   = (lane < 16) ? 0 : 8;
  const int ln   = lane & 15;
  const int k0 = (wv >> 1) * 16, d0 = (wv & 1) * 16;
  v8f acc = {};
#pragma unroll
  for (int jj = 0; jj < LL; jj += 4) {
    v2f a, bf;
    a.x  = Stile[(jj + klo) * LL + k0 + ln];
    a.y  = Stile[(jj + klo + 1) * LL + k0 + ln];
    bf.x = Q5[(jj + klo) * DD + d0 + ln];
    bf.y = Q5[(jj + klo + 1) * DD + d0 + ln];
    acc = wmma4(a, bf, acc);
  }
#pragma unroll
  for (int r = 0; r < 8; ++r) {
    const int ktok = k0 + r + mo;
    atomicAdd(&out[(b * LL + ktok) * HID + h * DD + d0 + ln], acc[r]);
  }
}

// ---------------------------------------------------------------------------
extern "C" void kernel_launch(void* const* d_in, const int* in_sizes, int n_in,
                              void* d_out, int out_size, void* d_ws, size_t ws_size,
                              hipStream_t stream) {
  (void)in_sizes; (void)n_in; (void)ws_size;
  const float* hid  = (const float*)d_in[0];
  const float* mask = (const float*)d_in[1];
  const float* sm   = (const float*)d_in[2];
  const float* trip = (const float*)d_in[3];
  const float* Wq  = (const float*)d_in[4];  const float* bq  = (const float*)d_in[5];
  const float* Wk  = (const float*)d_in[6];  const float* bk  = (const float*)d_in[7];
  const float* Wv  = (const float*)d_in[8];  const float* bv  = (const float*)d_in[9];
  const float* Wsq = (const float*)d_in[10]; const float* bsq = (const float*)d_in[11];
  const float* Wsv = (const float*)d_in[12]; const float* bsv = (const float*)d_in[13];
  const float* Wtq = (const float*)d_in[14]; const float* btq = (const float*)d_in[15];
  const float* Wtk = (const float*)d_in[16]; const float* btk = (const float*)d_in[17];
  float* out = (float*)d_out;

  float* ws   = (float*)d_ws;
  float* qo   = ws;                 // B*H*L*D   = 131072
  float* ko   = qo + 131072;
  float* vo   = ko + 131072;
  float* gko  = vo + 131072;
  float* sqo  = gko + 131072;       // B*L*L*D   = 1048576
  float* svo  = sqo + 1048576;
  float* c2o  = svo + 1048576;      // B*H*L     = 4096
  float* Ssum = c2o + 4096;         // B*H*L     = 4096

  (void)hipMemsetAsync(Ssum, 0, (size_t)BB * HH * LL * sizeof(float), stream);
  (void)hipMemsetAsync(out, 0, (size_t)out_size * sizeof(float), stream);

  k_proj_qkv<<<BB * LL, 256, 0, stream>>>(hid, Wq, bq, Wk, bk, Wv, bv, Wtk, btk,
                                          qo, ko, vo, gko, c2o);
  k_proj_sqsv<<<BB * LL * LL, 64, 0, stream>>>(sm, Wsq, bsq, Wsv, bsv, sqo, svo);

  dim3 grid(HH, LL, BB);  // h fastest -> sibling heads share triple slice in L2
  k_pass_sum<<<grid, 256, 0, stream>>>(qo, ko, gko, sqo, c2o, mask, trip, Wtq, btq, Ssum);
  k_pass_ctx<<<grid, 256, 0, stream>>>(qo, ko, vo, gko, sqo, svo, c2o, mask, trip,
                                       Wtq, btq, Ssum, out);
}